// GNNEncoder_70884140253627
// MI455X (gfx1250) — compile-verified
//
#include <hip/hip_runtime.h>
#include <hip/hip_bf16.h>

// ---------------------------------------------------------------------------
// GNN encoder for MI455X (gfx1250, wave32, WMMA).
// f32 end-to-end; GEMMs on V_WMMA_F32_16X16X4_F32 with double-buffered LDS
// tiles fed by the CDNA5 async-to-LDS path (ASYNCcnt).
// ---------------------------------------------------------------------------

typedef __attribute__((ext_vector_type(2))) float v2f;
typedef __attribute__((ext_vector_type(4))) float v4f;
typedef __attribute__((ext_vector_type(8))) float v8f;

#define BB   64
#define SS   1000
#define NN   (BB * SS)                // 64000 nodes
#define DD   128
#define HH   512
#define KNN  7                        // kappa + 2
#define NDEP 998                      // S - 2
#define EPB  (2 * NDEP + KNN * NDEP)  // 8982 edges / batch
#define ETOT (BB * EPB)               // 574848 edges total

#if defined(__HIP_DEVICE_COMPILE__) &&                                   \
    __has_builtin(__builtin_amdgcn_global_load_async_to_lds_b128) &&     \
    __has_builtin(__builtin_amdgcn_s_wait_asynccnt)
#define USE_ASYNC_LDS 1
#else
#define USE_ASYNC_LDS 0
#endif

#if USE_ASYNC_LDS
// Builtin arg0 type (from hipcc diagnostic): int __vector_size__(16) *
typedef int v4i_g __attribute__((__vector_size__(16)));
typedef __attribute__((address_space(3))) v4i_g* lds_v4i_ptr;

__device__ __forceinline__ v4i_g* async_src(const void* p) {
    return (v4i_g*)p;   // flat pointer to global buffer == global address
}
// Generic->LDS: low 32 bits of a flat shared pointer are the LDS byte offset.
__device__ __forceinline__ lds_v4i_ptr async_dst(void* p) {
    return (lds_v4i_ptr)(unsigned int)(unsigned long long)p;
}
#endif

// ---------------------------------------------------------------------------
// 1) Node embedding: x = inputs@Wc + bc + scor@Ws + bs   [NN x 128]
// ---------------------------------------------------------------------------
__global__ __launch_bounds__(128) void embed_kernel(
    const float* __restrict__ inputs, const float* __restrict__ scor,
    const float* __restrict__ Wc, const float* __restrict__ bc,
    const float* __restrict__ Ws, const float* __restrict__ bs,
    float* __restrict__ x)
{
    int n = blockIdx.x;
    int d = threadIdx.x;
    float i0 = inputs[2 * n + 0];
    float i1 = inputs[2 * n + 1];
    float s0 = scor[n];
    float v = bc[d] + bs[d];
    v = fmaf(s0, Ws[d], v);
    v = fmaf(i1, Wc[DD + d], v);
    v = fmaf(i0, Wc[d], v);
    x[(size_t)n * DD + d] = v;
}

// ---------------------------------------------------------------------------
// 2) kNN: 7 nearest by squared distance (monotonic w/ sqrt; ascending-j
//    insertion w/ strict < reproduces lax.top_k tie stability).
// ---------------------------------------------------------------------------
__global__ __launch_bounds__(256) void knn_kernel(
    const float* __restrict__ inputs, int* __restrict__ knn_idx)
{
    __shared__ float xs[SS];
    __shared__ float ys[SS];
    int b = blockIdx.y;
    const float* base = inputs + (size_t)b * SS * 2;
    for (int j = threadIdx.x; j < SS; j += blockDim.x) {
        xs[j] = base[2 * j + 0];
        ys[j] = base[2 * j + 1];
    }
    __syncthreads();

    int i = 2 + blockIdx.x * blockDim.x + threadIdx.x;
    if (i >= SS) return;
    float xi = xs[i], yi = ys[i];

    float bd[KNN]; int bj[KNN];
#pragma unroll
    for (int t = 0; t < KNN; ++t) { bd[t] = 3.4e38f; bj[t] = 0; }

    for (int j = 0; j < SS; ++j) {
        float dx = xi - xs[j], dy = yi - ys[j];
        float d2 = fmaf(dx, dx, dy * dy);
        if (j == i) d2 = 3.4e38f;
        if (d2 < bd[KNN - 1]) {
            bd[KNN - 1] = d2; bj[KNN - 1] = j;
#pragma unroll
            for (int t = KNN - 1; t > 0; --t) {
                if (bd[t] < bd[t - 1]) {
                    float td = bd[t]; bd[t] = bd[t - 1]; bd[t - 1] = td;
                    int   tj = bj[t]; bj[t] = bj[t - 1]; bj[t - 1] = tj;
                }
            }
        }
    }
    int* out = knn_idx + ((size_t)b * NDEP + (i - 2)) * KNN;
#pragma unroll
    for (int t = 0; t < KNN; ++t) out[t] = bj[t];
}

// ---------------------------------------------------------------------------
// 3) Graph: edges, degrees, symmetric GCN norm
// ---------------------------------------------------------------------------
__global__ __launch_bounds__(256) void deg_init_kernel(float* __restrict__ deg)
{
    int n = blockIdx.x * blockDim.x + threadIdx.x;
    if (n < NN) deg[n] = 1.0f;
}

__global__ __launch_bounds__(256) void build_edges_kernel(
    const int* __restrict__ knn_idx, int* __restrict__ esrc,
    int* __restrict__ edst, float* __restrict__ ew, float* __restrict__ deg)
{
    int e = blockIdx.x * blockDim.x + threadIdx.x;
    if (e >= ETOT) return;
    int b = e / EPB, le = e - b * EPB;
    int s, d; float w = 1.0f;
    if (le < NDEP) {                              // 0 -> i
        s = 0; d = 2 + le;
    } else if (le < 2 * NDEP) {                   // i -> 1
        s = 2 + (le - NDEP); d = 1;
    } else {                                      // kNN: i -> neighbor
        int k = le - 2 * NDEP;
        int i = k / KNN, nb = k - i * KNN;
        s = 2 + i;
        d = knn_idx[((size_t)b * NDEP + i) * KNN + nb];
        if (d < 2) w = 0.0f;
    }
    int off = b * SS;
    esrc[e] = off + s;
    edst[e] = off + d;
    ew[e]   = w;
    if (w != 0.0f) atomicAdd(&deg[off + d], 1.0f);
}

__global__ __launch_bounds__(256) void dinv_kernel(
    const float* __restrict__ deg, float* __restrict__ dinv,
    float* __restrict__ self_norm)
{
    int n = blockIdx.x * blockDim.x + threadIdx.x;
    if (n >= NN) return;
    float di = 1.0f / sqrtf(deg[n]);
    dinv[n] = di;
    self_norm[n] = di * di;
}

__global__ __launch_bounds__(256) void edge_norm_kernel(
    const int* __restrict__ esrc, const int* __restrict__ edst,
    const float* __restrict__ ew, const float* __restrict__ dinv,
    float* __restrict__ enorm)
{
    int e = blockIdx.x * blockDim.x + threadIdx.x;
    if (e >= ETOT) return;
    enorm[e] = dinv[esrc[e]] * dinv[edst[e]] * ew[e];
}

// ---------------------------------------------------------------------------
// 4) WMMA f32 GEMM, double-buffered LDS, async memory->LDS staging.
//    Hout = act(A) @ B ; Agg = Hout * self_norm[row] + bias[col]
//    BM=128 x BN=64, BK=32; 8 waves, each wave a 16x64 strip (4 x v8f).
//    RELU is a template parameter: layer-0 instantiation carries zero
//    activation VALU; ReLU layers get a single v_max per element.
// ---------------------------------------------------------------------------
#define BM 128
#define BN 64
#define BK 32
#define APAD 4
#define BPAD 8

template <int RELU>
__global__ __launch_bounds__(256) void gemm_wmma_f32_kernel(
    const float* __restrict__ A, const float* __restrict__ B,
    const float* __restrict__ bias, const float* __restrict__ self_norm,
    float* __restrict__ Hout, float* __restrict__ Agg,
    int K, int N)
{
    __shared__ float As[2][BM][BK + APAD];
    __shared__ float Bs[2][BK][BN + BPAD];

    int tid  = threadIdx.x;
    int wave = tid >> 5;
    int lane = tid & 31;
    int m0 = blockIdx.x * BM;
    int n0 = blockIdx.y * BN;
    int rowbase = wave * 16;

    v8f acc0 = {}, acc1 = {}, acc2 = {}, acc3 = {};

    // Per-thread staging coordinates (A: 4 x 16B, B: 2 x 16B per K-tile).
    int arr[4], arc[4], brr[2], brc[2];
#pragma unroll
    for (int it = 0; it < 4; ++it) {
        int li = tid + it * 256;
        arr[it] = li >> 3;
        arc[it] = (li & 7) << 2;
    }
#pragma unroll
    for (int it = 0; it < 2; ++it) {
        int li = tid + it * 256;
        brr[it] = li >> 4;
        brc[it] = (li & 15) << 2;
    }

#if USE_ASYNC_LDS
    auto stage = [&](int kk0, int bufi) {
#pragma unroll
        for (int it = 0; it < 4; ++it) {
            __builtin_amdgcn_global_load_async_to_lds_b128(
                async_src(A + (size_t)(m0 + arr[it]) * K + kk0 + arc[it]),
                async_dst(&As[bufi][arr[it]][arc[it]]), 0, 0);
        }
#pragma unroll
        for (int it = 0; it < 2; ++it) {
            __builtin_amdgcn_global_load_async_to_lds_b128(
                async_src(B + (size_t)(kk0 + brr[it]) * N + n0 + brc[it]),
                async_dst(&Bs[bufi][brr[it]][brc[it]]), 0, 0);
        }
    };
    stage(0, 0);
    __builtin_amdgcn_s_wait_asynccnt(0);
    __syncthreads();
#else
    v4f ra[4], rb[2];
    auto load_regs = [&](int kk0) {
#pragma unroll
        for (int it = 0; it < 4; ++it)
            ra[it] = *(const v4f*)(A + (size_t)(m0 + arr[it]) * K + kk0 + arc[it]);
#pragma unroll
        for (int it = 0; it < 2; ++it)
            rb[it] = *(const v4f*)(B + (size_t)(kk0 + brr[it]) * N + n0 + brc[it]);
    };
    auto store_regs = [&](int bufi) {
#pragma unroll
        for (int it = 0; it < 4; ++it)
            *(v4f*)&As[bufi][arr[it]][arc[it]] = ra[it];
#pragma unroll
        for (int it = 0; it < 2; ++it)
            *(v4f*)&Bs[bufi][brr[it]][brc[it]] = rb[it];
    };
    load_regs(0);
    store_regs(0);
    __syncthreads();
#endif

    int mloc = rowbase + (lane & 15);
    int kof  = (lane < 16) ? 0 : 2;      // per-lane address select, no divergence
    int nl   = lane & 15;

    int buf = 0;
    for (int k0 = 0; k0 < K; k0 += BK) {
        int more = (k0 + BK) < K;
#if USE_ASYNC_LDS
        if (more) stage(k0 + BK, buf ^ 1);   // DMA next tile during compute
#else
        if (more) load_regs(k0 + BK);        // issue loads before compute
#endif

#pragma unroll
        for (int kk = 0; kk < BK; kk += 4) {
            v2f a;
            a.x = As[buf][mloc][kk + kof];
            a.y = As[buf][mloc][kk + kof + 1];
            if (RELU) { a.x = fmaxf(a.x, 0.0f); a.y = fmaxf(a.y, 0.0f); }
            v2f bf;
            bf.x = Bs[buf][kk + kof][nl];        bf.y = Bs[buf][kk + kof + 1][nl];
            acc0 = __builtin_amdgcn_wmma_f32_16x16x4_f32(false, a, false, bf,
                                                         (short)0, acc0, false, false);
            bf.x = Bs[buf][kk + kof][nl + 16];   bf.y = Bs[buf][kk + kof + 1][nl + 16];
            acc1 = __builtin_amdgcn_wmma_f32_16x16x4_f32(false, a, false, bf,
                                                         (short)0, acc1, false, false);
            bf.x = Bs[buf][kk + kof][nl + 32];   bf.y = Bs[buf][kk + kof + 1][nl + 32];
            acc2 = __builtin_amdgcn_wmma_f32_16x16x4_f32(false, a, false, bf,
                                                         (short)0, acc2, false, false);
            bf.x = Bs[buf][kk + kof][nl + 48];   bf.y = Bs[buf][kk + kof + 1][nl + 48];
            acc3 = __builtin_amdgcn_wmma_f32_16x16x4_f32(false, a, false, bf,
                                                         (short)0, acc3, false, false);
        }

#if USE_ASYNC_LDS
        __builtin_amdgcn_s_wait_asynccnt(0);
#else
        if (more) store_regs(buf ^ 1);
#endif
        __syncthreads();
        buf ^= 1;
    }

    // Epilogue. C/D: VGPR j -> row j (lanes 0-15) / row j+8 (lanes 16-31).
    int mres = m0 + rowbase + ((lane < 16) ? 0 : 8);
    int nres = n0 + (lane & 15);
#pragma unroll
    for (int j = 0; j < 8; ++j) {
        int row = mres + j;
        float sn = self_norm[row];
        size_t base = (size_t)row * N;
        float v;
        v = acc0[j]; Hout[base + nres +  0] = v; Agg[base + nres +  0] = fmaf(v, sn, bias[nres +  0]);
        v = acc1[j]; Hout[base + nres + 16] = v; Agg[base + nres + 16] = fmaf(v, sn, bias[nres + 16]);
        v = acc2[j]; Hout[base + nres + 32] = v; Agg[base + nres + 32] = fmaf(v, sn, bias[nres + 32]);
        v = acc3[j]; Hout[base + nres + 48] = v; Agg[base + nres + 48] = fmaf(v, sn, bias[nres + 48]);
    }
}

// ---------------------------------------------------------------------------
// 5) Edge aggregation: Agg[dst] += h[src] * norm  (f32 atomics, float4 gather)
// ---------------------------------------------------------------------------
__global__ __launch_bounds__(256) void scatter_kernel(
    const int* __restrict__ esrc, const int* __restrict__ edst,
    const float* __restrict__ enorm, const float* __restrict__ Hin,
    float* __restrict__ Agg, int F)
{
    long long gid = (long long)blockIdx.x * blockDim.x + threadIdx.x;
    int f4 = F >> 2;
    long long tot = (long long)ETOT * f4;
    if (gid >= tot) return;
    int e = (int)(gid / f4);
    int c = (int)(gid - (long long)e * f4) << 2;
    float w = enorm[e];
    if (w == 0.0f) return;
    int s = esrc[e], d = edst[e];
    v4f hv = *(const v4f*)(Hin + (size_t)s * F + c);
    float* out = Agg + (size_t)d * F + c;
    atomicAdd(out + 0, hv.x * w);
    atomicAdd(out + 1, hv.y * w);
    atomicAdd(out + 2, hv.z * w);
    atomicAdd(out + 3, hv.w * w);
}

// ---------------------------------------------------------------------------
// Launch
// ---------------------------------------------------------------------------
extern "C" void kernel_launch(void* const* d_in, const int* in_sizes, int n_in,
                              void* d_out, int out_size, void* d_ws, size_t ws_size,
                              hipStream_t stream)
{
    const float* inputs = (const float*)d_in[0];
    const float* scor   = (const float*)d_in[1];
    const float* Wc     = (const float*)d_in[2];
    const float* bc     = (const float*)d_in[3];
    const float* Ws     = (const float*)d_in[4];
    const float* bs     = (const float*)d_in[5];
    const float* W0     = (const float*)d_in[6];
    const float* b0     = (const float*)d_in[7];
    const float* W1     = (const float*)d_in[8];
    const float* b1     = (const float*)d_in[9];
    const float* W2     = (const float*)d_in[10];
    const float* b2     = (const float*)d_in[11];
    float* out = (float*)d_out;

    char* wsb = (char*)d_ws;
    size_t off = 0;
    auto alloc = [&](size_t bytes) -> char* {
        char* p = wsb + off;
        off += (bytes + 255) & ~(size_t)255;
        return p;
    };
    float* x    = (float*)alloc((size_t)NN * DD * sizeof(float));
    float* h    = (float*)alloc((size_t)NN * HH * sizeof(float));
    float* agg0 = (float*)alloc((size_t)NN * HH * sizeof(float));
    float* agg1 = (float*)alloc((size_t)NN * HH * sizeof(float));
    int*   kidx = (int*)  alloc((size_t)BB * NDEP * KNN * sizeof(int));
    int*   esrc = (int*)  alloc((size_t)ETOT * sizeof(int));
    int*   edst = (int*)  alloc((size_t)ETOT * sizeof(int));
    float* ew   = (float*)alloc((size_t)ETOT * sizeof(float));
    float* enrm = (float*)alloc((size_t)ETOT * sizeof(float));
    float* deg  = (float*)alloc((size_t)NN * sizeof(float));
    float* dinv = (float*)alloc((size_t)NN * sizeof(float));
    float* snrm = (float*)alloc((size_t)NN * sizeof(float));

    embed_kernel<<<NN, DD, 0, stream>>>(inputs, scor, Wc, bc, Ws, bs, x);
    dim3 kg((NDEP + 255) / 256, BB);
    knn_kernel<<<kg, 256, 0, stream>>>(inputs, kidx);
    deg_init_kernel<<<(NN + 255) / 256, 256, 0, stream>>>(deg);
    build_edges_kernel<<<(ETOT + 255) / 256, 256, 0, stream>>>(kidx, esrc, edst, ew, deg);
    dinv_kernel<<<(NN + 255) / 256, 256, 0, stream>>>(deg, dinv, snrm);
    edge_norm_kernel<<<(ETOT + 255) / 256, 256, 0, stream>>>(esrc, edst, ew, dinv, enrm);

    // layer 0: h = x @ W0 ; agg0 = h*sn + b0 ; agg0 += scatter(h)
    gemm_wmma_f32_kernel<0><<<dim3(NN / BM, HH / BN), 256, 0, stream>>>(
        x, W0, b0, snrm, h, agg0, DD, HH);
    scatter_kernel<<<(unsigned)(((long long)ETOT * (HH / 4) + 255) / 256), 256, 0, stream>>>(
        esrc, edst, enrm, h, agg0, HH);

    // layer 1: h = relu(agg0) @ W1 ; agg1 = h*sn + b1 ; += scatter(h)
    gemm_wmma_f32_kernel<1><<<dim3(NN / BM, HH / BN), 256, 0, stream>>>(
        agg0, W1, b1, snrm, h, agg1, HH, HH);
    scatter_kernel<<<(unsigned)(((long long)ETOT * (HH / 4) + 255) / 256), 256, 0, stream>>>(
        esrc, edst, enrm, h, agg1, HH);

    // layer 2: h = relu(agg1) @ W2 ; out = h*sn + b2 ; += scatter(h)
    gemm_wmma_f32_kernel<1><<<dim3(NN / BM, DD / BN), 256, 0, stream>>>(
        agg1, W2, b2, snrm, h, out, HH, DD);
    scatter_kernel<<<(unsigned)(((long long)ETOT * (DD / 4) + 255) / 256), 256, 0, stream>>>(
        esrc, edst, enrm, h, out, DD);
}